// TopoAwareQuantumLayer_39238821216466
// MI455X (gfx1250) — compile-verified
//
#include <hip/hip_runtime.h>
#include <math.h>

#define NQB 12
#define DIMS 4096
#define BATCH 256
#define HID 256
#define NE 12

typedef float v2f __attribute__((ext_vector_type(2)));
typedef float v8f __attribute__((ext_vector_type(8)));

// ---------- complex helpers ----------
__device__ __forceinline__ float2 cmul(float2 a, float2 b) {
    return make_float2(a.x * b.x - a.y * b.y, a.x * b.y + a.y * b.x);
}
__device__ __forceinline__ float2 cadd(float2 a, float2 b) {
    return make_float2(a.x + b.x, a.y + b.y);
}

struct G2 { float2 a, b, c, d; };  // [[a,b],[c,d]]

__device__ __forceinline__ G2 gmul(G2 P, G2 Q) {
    G2 r;
    r.a = cadd(cmul(P.a, Q.a), cmul(P.b, Q.c));
    r.b = cadd(cmul(P.a, Q.b), cmul(P.b, Q.d));
    r.c = cadd(cmul(P.c, Q.a), cmul(P.d, Q.c));
    r.d = cadd(cmul(P.c, Q.b), cmul(P.d, Q.d));
    return r;
}
__device__ __forceinline__ G2 mk_rx(float t) {
    float c = cosf(0.5f * t), s = sinf(0.5f * t);
    G2 g; g.a = make_float2(c, 0.f); g.b = make_float2(0.f, -s);
    g.c = make_float2(0.f, -s);     g.d = make_float2(c, 0.f);
    return g;
}
__device__ __forceinline__ G2 mk_ry(float t) {
    float c = cosf(0.5f * t), s = sinf(0.5f * t);
    G2 g; g.a = make_float2(c, 0.f); g.b = make_float2(-s, 0.f);
    g.c = make_float2(s, 0.f);      g.d = make_float2(c, 0.f);
    return g;
}
__device__ __forceinline__ G2 mk_rz(float t) {
    float c = cosf(0.5f * t), s = sinf(0.5f * t);
    G2 g; g.a = make_float2(c, -s); g.b = make_float2(0.f, 0.f);
    g.c = make_float2(0.f, 0.f);    g.d = make_float2(c, s);
    return g;
}
__device__ __forceinline__ float gelu_exact(float v) {
    return 0.5f * v * (1.f + erff(v * 0.70710678118654752f));
}
__device__ __forceinline__ float sigmoidf(float v) { return 1.f / (1.f + expf(-v)); }

// ---------- kernel 0: q0 = tanh(x[256,256] @ Wq[256,12] + bq) via WMMA f32 16x16x4 ----------
// One wave per 16-row tile; N padded 12->16 by branchless zero-masking (exact).
// No divergent control flow in the K loop: EXEC stays all-ones for every WMMA.
__global__ __launch_bounds__(32) void q0_wmma_kernel(
    const float* __restrict__ x, const float* __restrict__ Wq, const float* __restrict__ bq,
    float* __restrict__ q0)
{
    const int l = threadIdx.x;
    const int mt = blockIdx.x * 16;
    const int half = l >> 4, lo = l & 15;
    const bool nvalid = (lo < NQB);
    const int lw = nvalid ? lo : 0;          // clamped in-bounds column
    const float msk = nvalid ? 1.f : 0.f;    // zero-mask for padded columns

    v8f acc = {};
    #pragma unroll 8
    for (int k0 = 0; k0 < HID; k0 += 4) {
        const int ka = k0 + 2 * half;   // lanes 0-15: K={k0,k0+1}; lanes 16-31: K={k0+2,k0+3}
        v2f a, bm;
        a.x  = x[(mt + lo) * HID + ka];
        a.y  = x[(mt + lo) * HID + ka + 1];
        bm.x = Wq[ka * NQB + lw] * msk;
        bm.y = Wq[(ka + 1) * NQB + lw] * msk;
        acc = __builtin_amdgcn_wmma_f32_16x16x4_f32(
            false, a, false, bm, (short)0, acc, false, false);
    }
    if (nvalid) {
        const float bqv = bq[lo];
        #pragma unroll
        for (int r = 0; r < 8; ++r) {
            const int m = mt + r + 8 * half;
            q0[m * NQB + lo] = tanhf(acc[r] + bqv);
        }
    }
}

// ---------- kernel 1: remaining classical prep MLPs (per batch row) ----------
__global__ __launch_bounds__(64) void prep_kernel(
    const float* __restrict__ topo_node, const float* __restrict__ topo_graph,
    const float* __restrict__ q0_in,
    const float* __restrict__ We1, const float* __restrict__ be1,
    const float* __restrict__ We2, const float* __restrict__ be2,
    const float* __restrict__ Wi1, const float* __restrict__ bi1,
    const float* __restrict__ Wi2, const float* __restrict__ bi2,
    const float* __restrict__ Wi3, const float* __restrict__ bi3,
    const float* __restrict__ Wg1, const float* __restrict__ bg1,
    const float* __restrict__ Wg2, const float* __restrict__ bg2,
    float* __restrict__ qin_out, float* __restrict__ imod_out, float* __restrict__ gate_out)
{
    const int b = blockIdx.x, t = threadIdx.x;  // 64 threads
    __shared__ float tg22[22], tnm[21], hA[64], hB[64], e36[36], q0s[NQB], g32[32];

    if (t < 22) tg22[t] = topo_graph[b * 22 + t];
    if (t < 21) {
        float s = 0.f;
        for (int v = 0; v < NQB; ++v) s += topo_node[(b * NQB + v) * 21 + t];
        tnm[t] = s * (1.f / 12.f);
    }
    if (t < NQB) q0s[t] = q0_in[b * NQB + t];
    __syncthreads();

    // hA = gelu(tnm@We1 + be1)
    {
        float s = be1[t];
        for (int u = 0; u < 21; ++u) s += tnm[u] * We1[u * 64 + t];
        hA[t] = gelu_exact(s);
    }
    __syncthreads();

    if (t < 36) {
        float s = be2[t];
        for (int u = 0; u < 64; ++u) s += hA[u] * We2[u * 36 + t];
        e36[t] = tanhf(s);
    }
    __syncthreads();

    // q_in and first imod layer
    if (t < NQB) {
        float em = (e36[t * 3] + e36[t * 3 + 1] + e36[t * 3 + 2]) * (1.f / 3.f);
        qin_out[b * NQB + t] = q0s[t] * (1.f + 0.5f * em);
    }
    {
        float s = bi1[t];
        for (int u = 0; u < 22; ++u) s += tg22[u] * Wi1[u * 64 + t];
        hB[t] = gelu_exact(s);
    }
    __syncthreads();

    // second imod layer + gate hidden layer
    {
        float s = bi2[t];
        for (int u = 0; u < 64; ++u) s += hB[u] * Wi2[u * 64 + t];
        hA[t] = gelu_exact(s);
    }
    if (t < 32) {
        float s = bg1[t];
        for (int u = 0; u < 22; ++u) s += tg22[u] * Wg1[u * 32 + t];
        g32[t] = gelu_exact(s);
    }
    __syncthreads();

    if (t < 36) {
        float s = bi3[t];
        for (int u = 0; u < 64; ++u) s += hA[u] * Wi3[u * 36 + t];
        imod_out[b * 36 + t] = tanhf(s);
    }
    if (t == 0) {
        float s = bg2[0];
        for (int u = 0; u < 32; ++u) s += g32[u] * Wg2[u];
        gate_out[b] = sigmoidf(s);
    }
}

// ---------- kernel 2: statevector circuit, state resident in LDS ----------
__device__ __forceinline__ void apply1(float2* st, G2 g, int q, int tid) {
    const int sh = (NQB - 1) - q, str = 1 << sh;
    #pragma unroll
    for (int p = 0; p < 8; ++p) {
        int pid = tid + p * 256;                       // 2048 pairs
        int i0 = ((pid >> sh) << (sh + 1)) | (pid & (str - 1));
        int i1 = i0 | str;
        float2 a = st[i0], bb = st[i1];
        float2 na = cadd(cmul(g.a, a), cmul(g.b, bb));
        float2 nb = cadd(cmul(g.c, a), cmul(g.d, bb));
        st[i0] = na; st[i1] = nb;
    }
}

__global__ __launch_bounds__(256) void circuit_kernel(
    const float* __restrict__ qin, const float* __restrict__ imod,
    const float* __restrict__ eta,
    const float* __restrict__ ew_zz, const float* __restrict__ ew_xx, const float* __restrict__ ew_yy,
    const float* __restrict__ nb_z, const float* __restrict__ nb_x, const float* __restrict__ nb_y,
    const float* __restrict__ thetas, float* __restrict__ qc)
{
    __shared__ float2 st[DIMS];  // 32 KB statevector
    const int b = blockIdx.x, tid = threadIdx.x;

    #pragma unroll
    for (int a = 0; a < 16; ++a) {
        int idx = tid + a * 256;
        st[idx] = make_float2(idx == 0 ? 1.f : 0.f, 0.f);
    }
    __syncthreads();

    for (int k = 0; k < 3; ++k) {
        // ---- data encoding: k0 'full'=RZ*RY*RX, k1 'xy'=RY*RX, k2 'yz'=RZ*RY ----
        for (int v = 0; v < NQB; ++v) {
            float a = qin[b * NQB + v];
            G2 g;
            if (k == 0)      g = gmul(mk_rz(a), gmul(mk_ry(a), mk_rx(a)));
            else if (k == 1) g = gmul(mk_ry(a), mk_rx(a));
            else             g = gmul(mk_rz(a), mk_ry(a));
            apply1(st, g, v, tid);
            __syncthreads();
        }
        // ---- Ising edge gates ----
        const float et = eta[k];
        const float* ew = (k == 0) ? ew_zz : ((k == 1) ? ew_xx : ew_yy);
        for (int e = 0; e < NE; ++e) {
            const int qi = e, qj = (e + 1) % NQB;
            const float ang = et * ew[e] * (1.f + imod[b * 36 + k * NE + e]);
            const float c = cosf(0.5f * ang), s = sinf(0.5f * ang);
            const int shi = (NQB - 1) - qi, shj = (NQB - 1) - qj;
            const int stri = 1 << shi, strj = 1 << shj;
            if (k == 0) {  // ZZ: per-amplitude phase (c - i*s*sigma)
                #pragma unroll
                for (int a = 0; a < 16; ++a) {
                    int idx = tid + a * 256;
                    int bi = (idx >> shi) & 1, bj = (idx >> shj) & 1;
                    float t = (bi == bj) ? s : -s;
                    float2 v = st[idx];
                    st[idx] = make_float2(c * v.x + t * v.y, c * v.y - t * v.x);
                }
            } else {  // XX / YY: pair = flip both edge bits
                #pragma unroll
                for (int p = 0; p < 8; ++p) {
                    int pid = tid + p * 256;  // enumerate indices with bit qi == 0
                    int i0 = ((pid >> shi) << (shi + 1)) | (pid & (stri - 1));
                    int i1 = i0 ^ stri ^ strj;
                    int bj0 = (i0 >> shj) & 1;
                    float pp = (k == 1) ? 1.f : ((bj0 == 0) ? -1.f : 1.f);  // XX: +1; YY: -1 if bits equal
                    float ps = pp * s;
                    float2 v = st[i0], w = st[i1];
                    st[i0] = make_float2(c * v.x + ps * w.y, c * v.y - ps * w.x);
                    st[i1] = make_float2(c * w.x + ps * v.y, c * w.y - ps * v.x);
                }
            }
            __syncthreads();
        }
        // ---- fused (trainable RZ*RY*RX) * (bias rotation) per qubit ----
        float bang = et * ((k == 0) ? nb_z[0] : ((k == 1) ? nb_x[0] : nb_y[0]));
        G2 bias = (k == 0) ? mk_rz(bang) : ((k == 1) ? mk_rx(bang) : mk_ry(bang));
        for (int v = 0; v < NQB; ++v) {
            const float* th = thetas + (k * NQB + v) * 3;  // R=1
            G2 tgm = gmul(mk_rz(th[2]), gmul(mk_ry(th[1]), mk_rx(th[0])));
            apply1(st, gmul(tgm, bias), v, tid);
            __syncthreads();
        }
        // ---- CNOT ladder (skipped for last block, R=1) ----
        if (k != 2) {
            for (int q = 0; q < NQB - 1; ++q) {
                const int pos = (NQB - 1) - q;     // control bit
                const int tpos = pos - 1;          // target bit
                #pragma unroll
                for (int p = 0; p < 4; ++p) {
                    int pid = tid + p * 256;       // 1024 swap pairs
                    int low = pid & ((1 << tpos) - 1);
                    int high = pid >> tpos;
                    int idx = (high << (tpos + 2)) | (1 << pos) | low;  // control=1, target=0
                    int pidx = idx | (1 << tpos);
                    float2 a = st[idx];
                    st[idx] = st[pidx];
                    st[pidx] = a;
                }
                __syncthreads();
            }
        }
    }

    // ---- <Z_v> = sum_idx sign_v(idx) * |amp|^2 ----
    float loc[NQB];
    #pragma unroll
    for (int q = 0; q < NQB; ++q) loc[q] = 0.f;
    #pragma unroll
    for (int a = 0; a < 16; ++a) {
        int idx = tid + a * 256;
        float2 v = st[idx];
        float pr = v.x * v.x + v.y * v.y;
        #pragma unroll
        for (int q = 0; q < NQB; ++q)
            loc[q] += ((idx >> ((NQB - 1) - q)) & 1) ? -pr : pr;
    }
    __syncthreads();                       // done reading state; reuse LDS
    float* red = (float*)st;               // 256*12 floats = 12 KB
    #pragma unroll
    for (int q = 0; q < NQB; ++q) red[tid * NQB + q] = loc[q];
    __syncthreads();
    for (int sdist = 128; sdist > 0; sdist >>= 1) {
        if (tid < sdist) {
            #pragma unroll
            for (int q = 0; q < NQB; ++q)
                red[tid * NQB + q] += red[(tid + sdist) * NQB + q];
        }
        __syncthreads();
    }
    if (tid < NQB) qc[b * NQB + tid] = red[tid];
}

// ---------- kernel 3: gated blend ----------
__global__ __launch_bounds__(256) void mix_kernel(
    const float* __restrict__ q0, const float* __restrict__ qc, const float* __restrict__ gate,
    const float* __restrict__ Wfix, const float* __restrict__ bfix, float* __restrict__ mixed)
{
    int gid = blockIdx.x * 256 + threadIdx.x;  // 3072
    if (gid >= BATCH * NQB) return;
    int b = gid / NQB, v = gid % NQB;
    float fix = bfix[v];
    for (int u = 0; u < NQB; ++u) fix += q0[b * NQB + u] * Wfix[u * NQB + v];
    float g = gate[b];
    mixed[gid] = g * qc[gid] + (1.f - g) * fix;
}

// ---------- kernel 4: out = mixed[256,12] @ Wout[12,256] + bout via WMMA f32 16x16x4 ----------
__global__ __launch_bounds__(32) void out_wmma_kernel(
    const float* __restrict__ mixed, const float* __restrict__ Wout,
    const float* __restrict__ bout, float* __restrict__ out)
{
    const int l = threadIdx.x;              // one wave per 16x16 tile
    const int mt = blockIdx.x * 16, nt = blockIdx.y * 16;
    const int half = l >> 4, lo = l & 15;

    v8f acc = {};
    #pragma unroll
    for (int k0 = 0; k0 < 12; k0 += 4) {
        const int ka = k0 + 2 * half;       // lanes 0-15: K={k0,k0+1}; lanes 16-31: K={k0+2,k0+3}
        v2f a, bm;
        a.x  = mixed[(mt + lo) * NQB + ka];
        a.y  = mixed[(mt + lo) * NQB + ka + 1];
        bm.x = Wout[ka * HID + nt + lo];
        bm.y = Wout[(ka + 1) * HID + nt + lo];
        acc = __builtin_amdgcn_wmma_f32_16x16x4_f32(
            false, a, false, bm, (short)0, acc, false, false);
    }
    const float bo = bout[nt + lo];
    #pragma unroll
    for (int r = 0; r < 8; ++r)
        out[(mt + r + 8 * half) * HID + nt + lo] = acc[r] + bo;  // D: VGPR r -> M=r+8*(lane>=16)
}

// ---------- launcher ----------
extern "C" void kernel_launch(void* const* d_in, const int* in_sizes, int n_in,
                              void* d_out, int out_size, void* d_ws, size_t ws_size,
                              hipStream_t stream) {
    const float* x          = (const float*)d_in[0];
    const float* topo_node  = (const float*)d_in[1];
    const float* topo_graph = (const float*)d_in[2];
    const float* Wq  = (const float*)d_in[3];  const float* bq  = (const float*)d_in[4];
    const float* We1 = (const float*)d_in[5];  const float* be1 = (const float*)d_in[6];
    const float* We2 = (const float*)d_in[7];  const float* be2 = (const float*)d_in[8];
    const float* Wi1 = (const float*)d_in[9];  const float* bi1 = (const float*)d_in[10];
    const float* Wi2 = (const float*)d_in[11]; const float* bi2 = (const float*)d_in[12];
    const float* Wi3 = (const float*)d_in[13]; const float* bi3 = (const float*)d_in[14];
    const float* Wg1 = (const float*)d_in[15]; const float* bg1 = (const float*)d_in[16];
    const float* Wg2 = (const float*)d_in[17]; const float* bg2 = (const float*)d_in[18];
    const float* Wfix = (const float*)d_in[19]; const float* bfix = (const float*)d_in[20];
    const float* Wout = (const float*)d_in[21]; const float* bout = (const float*)d_in[22];
    const float* eta   = (const float*)d_in[23];
    const float* ew_zz = (const float*)d_in[24];
    const float* ew_xx = (const float*)d_in[25];
    const float* ew_yy = (const float*)d_in[26];
    const float* nb_z  = (const float*)d_in[27];
    const float* nb_x  = (const float*)d_in[28];
    const float* nb_y  = (const float*)d_in[29];
    const float* thetas = (const float*)d_in[30];

    float* ws    = (float*)d_ws;
    float* q0    = ws;            // 3072
    float* qin   = ws + 3072;     // 3072
    float* imod  = ws + 6144;     // 9216
    float* gate  = ws + 15360;    // 256
    float* qc    = ws + 15616;    // 3072
    float* mixed = ws + 18688;    // 3072
    float* outp  = (float*)d_out;

    q0_wmma_kernel<<<BATCH / 16, 32, 0, stream>>>(x, Wq, bq, q0);

    prep_kernel<<<BATCH, 64, 0, stream>>>(
        topo_node, topo_graph, q0, We1, be1, We2, be2,
        Wi1, bi1, Wi2, bi2, Wi3, bi3, Wg1, bg1, Wg2, bg2,
        qin, imod, gate);

    circuit_kernel<<<BATCH, 256, 0, stream>>>(
        qin, imod, eta, ew_zz, ew_xx, ew_yy, nb_z, nb_x, nb_y, thetas, qc);

    mix_kernel<<<(BATCH * NQB + 255) / 256, 256, 0, stream>>>(
        q0, qc, gate, Wfix, bfix, mixed);

    out_wmma_kernel<<<dim3(BATCH / 16, HID / 16), 32, 0, stream>>>(
        mixed, Wout, bout, outp);
}